// CoordsToNRF_824633721642
// MI455X (gfx1250) — compile-verified
//
#include <hip/hip_runtime.h>

// ---------------------------------------------------------------------------
// CoordsToNRF on MI455X (gfx1250, wave32).
//
// nrf[b,p] = atom_nc[b,p] * (AU2KCALMOLA / MAX_NRF) / d2(i,j),
// p = i*(i-1)/2 + j (strict lower triangle, row-major), d2 = |xi - xj|^2.
//
// d2 is computed 16x16-tile-at-a-time on the matrix pipe via
// V_WMMA_F32_16X16X4_F32 using the augmented-coordinate trick:
//   A row i  = [ xi.x,  xi.y,  xi.z, |xi|^2 ]          (16x4)
//   B col j  = [-2xj.x, -2xj.y, -2xj.z, 1   ]          (4x16)
//   C[m][n]  = |xj(n)|^2                               (16x16)
//   D = A*B + C = |xi|^2 - 2 xi.xj + |xj|^2 = d2
//
// Roofline: mandatory traffic = atom_nc read + nrf write = 134 MB
// -> ~5.7 us at 23.3 TB/s. coords (768 KB) is L2-resident. Tile coords are
// forced uniform via readfirstlane so tile decode is SALU, the triangle
// branch is scalar, and per-element addressing is saddr + 32-bit voffset.
// Tail is ~8 VALU/element (raw v_sqrt, v_rcp + 1 NR step).
// ---------------------------------------------------------------------------

typedef float v2f __attribute__((ext_vector_type(2)));
typedef float v8f __attribute__((ext_vector_type(8)));

#define N_ATOMS   512
#define N_PAIRS   130816               // 512*511/2
#define BATCH     128
#define TILE      16
#define NT        (N_ATOMS / TILE)     // 32 tile-rows
#define N_TILES   (NT * (NT + 1) / 2)  // 528 lower+diag tile pairs
#define WAVES_PER_BLOCK 8
#define THREADS   (WAVES_PER_BLOCK * 32)

// AU2KCALMOLA / MAX_NRF, folded at compile time.
#define NRF_COEF ((float)(627.5095 * 0.529177 / 100.0))

// v_rcp_f32 + one Newton-Raphson step (~0.5-1 ulp, no IEEE div expansion).
static __device__ __forceinline__ float fast_recip(float x) {
    float r = __builtin_amdgcn_rcpf(x);
    return __builtin_fmaf(r, __builtin_fmaf(-x, r, 1.0f), r);
}

__global__ __launch_bounds__(THREADS)
void CoordsToNRF_824633721642_kernel(const float* __restrict__ coords,
                                     const float* __restrict__ atom_nc,
                                     float* __restrict__ out) {
    const unsigned lane   = threadIdx.x & 31u;
    const unsigned n      = lane & 15u;          // column within tile
    const bool     laneHi = lane >= 16u;

    // One wave per (batch, tile). w is identical across the wave; force it
    // into an SGPR so all tile math is SALU and branches are scalar.
    const int w = __builtin_amdgcn_readfirstlane(
        (int)blockIdx.x * WAVES_PER_BLOCK + (int)(threadIdx.x >> 5));
    const int b = w / N_TILES;
    const int t = w - b * N_TILES;               // 0 .. 527

    // Decode tile (ti, tj), tj <= ti, from t = ti*(ti+1)/2 + tj. sqrt estimate
    // is off by at most 1; fix up branchlessly.
    int ti = (int)((__builtin_amdgcn_sqrtf((float)(8 * t + 1)) - 1.0f) * 0.5f);
    ti -= (ti * (ti + 1) / 2 > t) ? 1 : 0;
    ti += ((ti + 1) * (ti + 2) / 2 <= t) ? 1 : 0;
    ti = __builtin_amdgcn_readfirstlane(ti);
    const int tj = t - ti * (ti + 1) / 2;
    const int i0 = ti * TILE;
    const int j0 = tj * TILE;

    const float* __restrict__ cb = coords + (size_t)b * (N_ATOMS * 3);

    // --- Gather this tile's 16 row-atoms and 16 col-atoms (L2-resident) ----
    const int ai = i0 + (int)n;                  // row atom for this lane
    const float ax = cb[ai * 3 + 0];
    const float ay = cb[ai * 3 + 1];
    const float az = cb[ai * 3 + 2];
    const float an = __builtin_fmaf(ax, ax, __builtin_fmaf(ay, ay, az * az));

    const int bj = j0 + (int)n;                  // col atom for this lane
    const float bx = cb[bj * 3 + 0];
    const float by = cb[bj * 3 + 1];
    const float bz = cb[bj * 3 + 2];
    const float bn = __builtin_fmaf(bx, bx, __builtin_fmaf(by, by, bz * bz));

    // --- Build WMMA fragments (f32 16x16x4 layout) -------------------------
    // A 16x4: lanes 0-15 hold {K0,K1}, lanes 16-31 hold {K2,K3}, M = lane&15.
    v2f afrag;
    afrag.x = laneHi ? az : ax;
    afrag.y = laneHi ? an : ay;
    // B 4x16 mirrors A with M->N.
    v2f bfrag;
    bfrag.x = laneHi ? (-2.0f * bz) : (-2.0f * bx);
    bfrag.y = laneHi ? 1.0f         : (-2.0f * by);
    // C 16x16: C[m][n] = |xj(n)|^2 -> same value in all 8 C VGPRs per lane.
    v8f c = {bn, bn, bn, bn, bn, bn, bn, bn};

    // D[m][n] = d2 between atom i0+m and atom j0+n.
    v8f d2 = __builtin_amdgcn_wmma_f32_16x16x4_f32(
        /*neg_a=*/false, afrag, /*neg_b=*/false, bfrag,
        /*c_mod=*/(short)0, c, /*reuse_a=*/false, /*reuse_b=*/false);

    // --- Elementwise tail: nc * COEF / (sqrt(d2))^2 ------------------------
    // C/D layout: VGPR v -> row M=v (lanes 0-15) or M=v+8 (lanes 16-31).
    // Batch bases are scalar (b is SGPR) -> saddr + 32-bit voffset addressing.
    const float* __restrict__ ncB  = atom_nc + (size_t)b * N_PAIRS;
    float*       __restrict__ outB = out     + (size_t)b * N_PAIRS;
    const unsigned jcol = (unsigned)(j0 + (int)n);

    unsigned irow0 = (unsigned)(i0 + (laneHi ? 8 : 0));

    // Per-row pair offsets: p(row+1) = p(row) + row (strength-reduced).
    unsigned pv[8];
    {
        unsigned irow = irow0;
        unsigned p = irow * (irow - 1u) / 2u + jcol;
#pragma unroll
        for (int v = 0; v < 8; ++v) { pv[v] = p; p += irow; ++irow; }
    }

    if (i0 > j0) {
        // Strictly-below-diagonal tile: all 256 elements valid, no predication.
        float anc[8];
#pragma unroll
        for (int v = 0; v < 8; ++v) anc[v] = ncB[pv[v]];   // clause of 8 loads
#pragma unroll
        for (int v = 0; v < 8; ++v) {
            const float r = __builtin_amdgcn_sqrtf(d2[v]); // match sqrt->square
            outB[pv[v]] = (anc[v] * NRF_COEF) * fast_recip(r * r);
        }
    } else {
        // Diagonal tile (i0 == j0): keep only strict lower triangle.
#pragma unroll
        for (int v = 0; v < 8; ++v) {
            const unsigned irow = irow0 + (unsigned)v;
            if (irow > jcol) {
                const float anc = ncB[pv[v]];
                const float r   = __builtin_amdgcn_sqrtf(d2[v]);
                outB[pv[v]] = (anc * NRF_COEF) * fast_recip(r * r);
            }
        }
    }
}

extern "C" void kernel_launch(void* const* d_in, const int* in_sizes, int n_in,
                              void* d_out, int out_size, void* d_ws, size_t ws_size,
                              hipStream_t stream) {
    (void)in_sizes; (void)n_in; (void)d_ws; (void)ws_size; (void)out_size;
    const float* coords  = (const float*)d_in[0];   // [128, 512, 3] f32
    const float* atom_nc = (const float*)d_in[1];   // [128, 130816] f32
    float* out = (float*)d_out;                     // [128, 130816] f32

    const int total_waves = BATCH * N_TILES;        // 67584
    const int grid = total_waves / WAVES_PER_BLOCK; // 8448, exact
    CoordsToNRF_824633721642_kernel<<<grid, THREADS, 0, stream>>>(coords, atom_nc, out);
}